// FancyNet_25786983645203
// MI455X (gfx1250) — compile-verified
//
#include <hip/hip_runtime.h>
#include <hip/hip_bf16.h>
#include <math.h>

typedef __attribute__((ext_vector_type(2))) float v2f;
typedef __attribute__((ext_vector_type(4))) float v4f;
typedef __attribute__((ext_vector_type(8))) float v8f;

#define BN_EPS 1e-5f

static __device__ __forceinline__ float elu1(float x) {
  return x > 0.f ? x : (expf(x) - 1.f);
}

// ---------------- utility kernels ----------------

__global__ void k_zero(float* __restrict__ p, long long n) {
  long long i = (long long)blockIdx.x * blockDim.x + threadIdx.x;
  if (i < n) p[i] = 0.f;
}

__global__ void k_copy(float* __restrict__ d, const float* __restrict__ s, long long n) {
  long long i = (long long)blockIdx.x * blockDim.x + threadIdx.x;
  if (i < n) d[i] = s[i];
}

// column sums / sum-of-squares: grid.x = cols, grid.y = row chunks
__global__ void k_colstats(const float* __restrict__ x, int rows, int cols,
                           float* __restrict__ sum, float* __restrict__ sumsq) {
  int c = blockIdx.x;
  float s = 0.f, q = 0.f;
  for (long long r = (long long)blockIdx.y * blockDim.x + threadIdx.x; r < rows;
       r += (long long)gridDim.y * blockDim.x) {
    float v = x[r * cols + c];
    s += v; q += v * v;
  }
  __shared__ float shs[256], shq[256];
  int t = threadIdx.x;
  shs[t] = s; shq[t] = q;
  __syncthreads();
  for (int o = 128; o > 0; o >>= 1) {
    if (t < o) { shs[t] += shs[t + o]; shq[t] += shq[t + o]; }
    __syncthreads();
  }
  if (t == 0) { atomicAdd(&sum[c], shs[0]); atomicAdd(&sumsq[c], shq[0]); }
}

// y = maybe_relu((x - m) * rsqrt(v + eps) * g + b), training-mode stats from sums
__global__ void k_bn_apply(const float* __restrict__ x, float* __restrict__ y,
                           const float* __restrict__ sum, const float* __restrict__ sumsq,
                           const float* __restrict__ g, const float* __restrict__ b,
                           int rows, int cols, int relu) {
  long long i = (long long)blockIdx.x * blockDim.x + threadIdx.x;
  if (i >= (long long)rows * cols) return;
  int c = (int)(i % cols);
  float inv = 1.f / (float)rows;
  float m = sum[c] * inv;
  float v = sumsq[c] * inv - m * m;
  float r = (x[i] - m) * rsqrtf(v + BN_EPS) * g[c] + b[c];
  if (relu) r = fmaxf(r, 0.f);
  y[i] = r;
}

// h_pre[n, 0:80] = bn1(feat[n, 0:6]) @ W_emb
__global__ void k_embed(const float* __restrict__ feat,
                        const float* __restrict__ fsum, const float* __restrict__ fsq,
                        const float* __restrict__ g1, const float* __restrict__ b1,
                        const float* __restrict__ Wemb, float* __restrict__ out, int rows) {
  long long gid = (long long)blockIdx.x * blockDim.x + threadIdx.x;
  if (gid >= (long long)rows * 80) return;
  int n = (int)(gid / 80), oc = (int)(gid % 80);
  float inv = 1.f / (float)rows;
  float acc = 0.f;
#pragma unroll
  for (int k = 0; k < 6; ++k) {
    float m = fsum[k] * inv;
    float v = fsq[k] * inv - m * m;
    float xb = (feat[(long long)n * 6 + k] - m) * rsqrtf(v + BN_EPS) * g1[k] + b1[k];
    acc += xb * Wemb[k * 80 + oc];
  }
  out[gid] = acc;
}

// agg = elu(h + agg), elementwise
__global__ void k_elu_add(const float* __restrict__ h, float* __restrict__ agg, long long n) {
  long long i = (long long)blockIdx.x * blockDim.x + threadIdx.x;
  if (i >= n) return;
  agg[i] = elu1(h[i] + agg[i]);
}

// dst[ii[e], :] += src[jj[e], :]  -- one thread per (edge, group-of-4-cols).
// COLS known at compile time (80): constant div, b128 source loads,
// 4x global_atomic_add_f32 per thread.
template <int COLS>
__global__ void k_scatter_add(float* __restrict__ dst, const float* __restrict__ src,
                              const int* __restrict__ ii, const int* __restrict__ jj,
                              unsigned E) {
  constexpr int G = COLS / 4;  // 4-col groups per row
  unsigned g = blockIdx.x * blockDim.x + threadIdx.x;
  if (g >= E * G) return;
  unsigned e = g / G;
  unsigned c4 = (g - e * G) * 4;
  long long srow = (long long)jj[e] * COLS + c4;
  long long drow = (long long)ii[e] * COLS + c4;
  v4f v = *(const v4f*)(src + srow);
  atomicAdd(&dst[drow + 0], v.x);
  atomicAdd(&dst[drow + 1], v.y);
  atomicAdd(&dst[drow + 2], v.z);
  atomicAdd(&dst[drow + 3], v.w);
}

// dst[r, :] = src[idx[r], :]
__global__ void k_gather_rows(float* __restrict__ dst, const float* __restrict__ src,
                              const int* __restrict__ idx, int rows, int cols) {
  long long g = (long long)blockIdx.x * blockDim.x + threadIdx.x;
  if (g >= (long long)rows * cols) return;
  int r = (int)(g / cols), c = (int)(g % cols);
  dst[g] = src[(long long)idx[r] * cols + c];
}

// dst[idx[r], :] = src[r, :]
__global__ void k_scatter_rows(float* __restrict__ dst, const float* __restrict__ src,
                               const int* __restrict__ idx, int rows, int cols) {
  long long g = (long long)blockIdx.x * blockDim.x + threadIdx.x;
  if (g >= (long long)rows * cols) return;
  int r = (int)(g / cols), c = (int)(g % cols);
  dst[(long long)idx[r] * cols + c] = src[g];
}

// ---------------- WMMA fp32 GEMM ----------------
// C[M,N] = act(A[M,K] @ B[K,N] + bias), row-major tight strides.
// One wave computes a 16 x (16*NT) strip: the A fragment is loaded once per
// k-step and reused across the NT column tiles (NT v_wmma_f32_16x16x4_f32 per
// k-step). M % 16 == 0, K % 4 == 0; N may be ragged inside the last tile
// (zero-fill masking via select, no EXEC divergence before the WMMAs).
template <int NT>
__global__ void k_gemm_wmma(const float* __restrict__ A, const float* __restrict__ B,
                            const float* __restrict__ bias, float* __restrict__ C,
                            int M, int N, int K, int act) {
  int wave = (int)(((long long)blockIdx.x * blockDim.x + threadIdx.x) >> 5);
  int lane = threadIdx.x & 31;
  int strips = M >> 4;
  if (wave >= strips) return;
  int m0 = wave << 4;
  int lr = lane & 15, half = lane >> 4;

  int col[NT];
  float cmask[NT];
  int colc[NT];
#pragma unroll
  for (int t = 0; t < NT; ++t) {
    col[t] = (t << 4) + lr;
    cmask[t] = (col[t] < N) ? 1.f : 0.f;
    colc[t] = (col[t] < N) ? col[t] : 0;
  }

  v8f acc[NT];
#pragma unroll
  for (int t = 0; t < NT; ++t) acc[t] = (v8f){};

  const float* arow = A + (long long)(m0 + lr) * K;
  for (int kb = 0; kb < K; kb += 4) {
    int ka = kb + 2 * half;  // lanes 0-15: K=kb,kb+1 ; lanes 16-31: K=kb+2,kb+3
    v2f a = *(const v2f*)(arow + ka);  // 8B-aligned (K even, ka even)
    const float* b0 = B + (long long)ka * N;
    const float* b1 = B + (long long)(ka + 1) * N;
#pragma unroll
    for (int t = 0; t < NT; ++t) {
      v2f b;
      b.x = b0[colc[t]] * cmask[t];
      b.y = b1[colc[t]] * cmask[t];
      acc[t] = __builtin_amdgcn_wmma_f32_16x16x4_f32(false, a, false, b,
                                                     (short)0, acc[t], false, false);
    }
  }

#pragma unroll
  for (int t = 0; t < NT; ++t) {
    if (col[t] < N) {
      float bv = bias ? bias[col[t]] : 0.f;
#pragma unroll
      for (int v = 0; v < 8; ++v) {  // C layout: VGPR v -> row m0 + v + 8*half
        int row = m0 + v + 8 * half;
        float r = acc[t][v] + bv;
        if (act) r = elu1(r);
        C[(long long)row * N + col[t]] = r;
      }
    }
  }
}

// ---------------- host orchestration ----------------

static inline unsigned gblk(long long n, int bs) {
  return (unsigned)((n + bs - 1) / bs);
}

extern "C" void kernel_launch(void* const* d_in, const int* in_sizes, int n_in,
                              void* d_out, int out_size, void* d_ws, size_t ws_size,
                              hipStream_t stream) {
  (void)in_sizes; (void)n_in; (void)out_size; (void)ws_size;

  const float* feat   = (const float*)d_in[0];
  const float* bn1_g  = (const float*)d_in[1];
  const float* bn1_b  = (const float*)d_in[2];
  const float* W_emb  = (const float*)d_in[3];
  const float* embn_g = (const float*)d_in[4];
  const float* embn_b = (const float*)d_in[5];
  const float* W_msg  = (const float*)d_in[6];
  const float* W1     = (const float*)d_in[7];
  const float* b1     = (const float*)d_in[8];
  const float* W2     = (const float*)d_in[9];
  const float* b2     = (const float*)d_in[10];
  const float* W3     = (const float*)d_in[11];
  const float* b3     = (const float*)d_in[12];
  const float* bn2_g  = (const float*)d_in[13];
  const float* bn2_b  = (const float*)d_in[14];
  const float* Wc     = (const float*)d_in[15];
  const int* iE[4] = {(const int*)d_in[16], (const int*)d_in[18],
                      (const int*)d_in[20], (const int*)d_in[22]};
  const int* jE[4] = {(const int*)d_in[17], (const int*)d_in[19],
                      (const int*)d_in[21], (const int*)d_in[23]};
  const int* up[4] = {(const int*)d_in[24], (const int*)d_in[25],
                      (const int*)d_in[26], (const int*)d_in[27]};

  const int N0 = 131072;
  const int NODESL[4] = {131072, 65536, 32768, 16384};
  const int NEXTL[4]  = {65536, 32768, 16384, 8192};

  const long long SZ80 = (long long)N0 * 80;
  float* W   = (float*)d_ws;
  float* H   = W;                        // current node features h
  float* MSG = W + SZ80;                 // node-transformed msgs / hud scratch
  float* AGG = W + 2 * SZ80;             // scatter target -> features
  float* L0  = W + 3 * SZ80;             // latest[0] [131072,80]
  float* L1  = L0 + SZ80;                // latest[1] [65536,80]
  float* L2  = L1 + (long long)65536 * 80;   // latest[2] [32768,80]
  float* STAT = L2 + (long long)32768 * 80;  // stats: sum @0, sumsq @128
  float* latest[4] = {L0, L1, L2, nullptr};  // latest[3] is never read

  const int BS = 256;

  // ---- BN1 stats on feat (6 cols), embed to 80, embn stats, ReLU ----
  k_zero<<<1, BS, 0, stream>>>(STAT, 256);
  k_colstats<<<dim3(6, 64), BS, 0, stream>>>(feat, N0, 6, STAT, STAT + 128);
  k_embed<<<gblk((long long)N0 * 80, BS), BS, 0, stream>>>(
      feat, STAT, STAT + 128, bn1_g, bn1_b, W_emb, MSG, N0);
  k_zero<<<1, BS, 0, stream>>>(STAT, 256);
  k_colstats<<<dim3(80, 64), BS, 0, stream>>>(MSG, N0, 80, STAT, STAT + 128);
  k_bn_apply<<<gblk((long long)N0 * 80, BS), BS, 0, stream>>>(
      MSG, H, STAT, STAT + 128, embn_g, embn_b, N0, 80, 1);

  // ---- 4 levels of message passing + push-down cascade ----
  for (int l = 0; l < 4; ++l) {
    int Nl = NODESL[l];
    unsigned El = 7u * (unsigned)Nl;
    long long nEl = (long long)Nl * 80;

    // msgN = h @ W_msg[l]   (node-level transform; (h[j])@W == (h@W)[j])
    int strips = Nl >> 4;
    k_gemm_wmma<5><<<(strips + 7) / 8, BS, 0, stream>>>(
        H, W_msg + (long long)l * 6400, nullptr, MSG, Nl, 80, 80, 0);

    // agg[i] += msgN[j] over edges; features = elu(h + agg) in AGG
    k_zero<<<gblk(nEl, BS), BS, 0, stream>>>(AGG, nEl);
    k_scatter_add<80><<<gblk((long long)El * 20, BS), BS, 0, stream>>>(
        AGG, MSG, iE[l], jE[l], El);
    k_elu_add<<<gblk(nEl, BS), BS, 0, stream>>>(H, AGG, nEl);

    // downsample: h_{l+1} = features[up_l]
    if (l < 3)
      k_gather_rows<<<gblk((long long)NEXTL[l] * 80, BS), BS, 0, stream>>>(
          H, AGG, up[l], NEXTL[l], 80);

    // push_info_down: cascade from level l to level 0
    const float* fd = AGG;
    for (int it = 0; it <= l; ++it) {
      int c = l - it;
      int Nc = NODESL[c];
      unsigned Ec = 7u * (unsigned)Nc;
      long long nEc = (long long)Nc * 80;

      // hud = segment_sum(fd[jj_c], ii_c) into MSG
      k_zero<<<gblk(nEc, BS), BS, 0, stream>>>(MSG, nEc);
      k_scatter_add<80><<<gblk((long long)Ec * 20, BS), BS, 0, stream>>>(
          MSG, fd, iE[c], jE[c], Ec);

      // latest[c] = hud  (only needed if a later level reads it: l < 3)
      if (l < 3 && latest[c])
        k_copy<<<gblk(nEc, BS), BS, 0, stream>>>(latest[c], MSG, nEc);

      // latest[c-1][up_{c-1}] = hud ; fd = latest[c-1]
      if (c > 0) {
        k_scatter_rows<<<gblk(nEc, BS), BS, 0, stream>>>(latest[c - 1], MSG, up[c - 1], Nc, 80);
        fd = latest[c - 1];
      }
    }
    // after l==3 inner loop, MSG holds h_up_down [N0, 80]
  }

  // ---- MLP head: 80->64 elu, 64->64 elu, 64->64 elu, BN, ->13 ----
  int strips0 = N0 >> 4;
  k_gemm_wmma<4><<<(strips0 + 7) / 8, BS, 0, stream>>>(MSG, W1, b1, H,   N0, 64, 80, 1);
  k_gemm_wmma<4><<<(strips0 + 7) / 8, BS, 0, stream>>>(H,   W2, b2, AGG, N0, 64, 64, 1);
  k_gemm_wmma<4><<<(strips0 + 7) / 8, BS, 0, stream>>>(AGG, W3, b3, H,   N0, 64, 64, 1);

  k_zero<<<1, BS, 0, stream>>>(STAT, 256);
  k_colstats<<<dim3(64, 64), BS, 0, stream>>>(H, N0, 64, STAT, STAT + 128);
  k_bn_apply<<<gblk((long long)N0 * 64, BS), BS, 0, stream>>>(
      H, AGG, STAT, STAT + 128, bn2_g, bn2_b, N0, 64, 0);

  k_gemm_wmma<1><<<(strips0 + 7) / 8, BS, 0, stream>>>(AGG, Wc, nullptr, (float*)d_out,
                                                       N0, 13, 64, 0);
}